// SharedVectorQuantizer_20615843021117
// MI455X (gfx1250) — compile-verified
//
#include <hip/hip_runtime.h>

// ---------------------------------------------------------------------------
// VQ quantizer for MI455X (gfx1250, wave32, WMMA).
//   tokens   = argmin_v ( ||e_v||^2 - 2 x.e_v )          (x_sq constant per row)
//   quant_st = codebook[tokens]                          (x + sg(q-x) == q)
//   vq_loss  = 1.25 * mean((q - x)^2)                    (emb == commit value)
// GEMM done with V_WMMA_F32_16X16X4_F32, K=128 -> 32 k-steps per 16x16 tile.
// ---------------------------------------------------------------------------

typedef __attribute__((ext_vector_type(2))) float v2f;
typedef __attribute__((ext_vector_type(8))) float v8f;

#define VQ_BN     65536    // 16 * 4096 tokens
#define VQ_V      4096     // codebook entries
#define VQ_D      128      // embed dim
#define VQ_CHUNKS 256      // VQ_V / 16
#define VQ_ELEMS  8388608  // VQ_BN * VQ_D

// --------------------------- ||e_v||^2 precompute ---------------------------
__global__ __launch_bounds__(256) void vq_esq_kernel(const float* __restrict__ cb,
                                                     float* __restrict__ esq) {
    int v = blockIdx.x * 256 + threadIdx.x;
    const float4* p = (const float4*)(cb + (size_t)v * VQ_D);
    float s = 0.0f;
#pragma unroll
    for (int i = 0; i < VQ_D / 4; ++i) {
        float4 q = p[i];
        s += q.x * q.x + q.y * q.y + q.z * q.z + q.w * q.w;
    }
    esq[v] = s;
}

// ------------------------------ main WMMA kernel ----------------------------
// grid: 512 blocks x 256 threads (8 waves). Wave w handles rows
// [blk*128 + w*16, +16). Codebook chunks (16 cols x 128 k) double-buffered in
// LDS; ||e||^2 folded into accumulator init; -2 folded into A registers.
// Single barrier per chunk: compute reads buf[cur] while stores target
// buf[1-cur]; cross-iteration hazards are covered by the end-of-iter barrier.
__global__ __launch_bounds__(256) void vq_argmin_kernel(const float* __restrict__ x,
                                                        const float* __restrict__ cb,
                                                        const float* __restrict__ esq,
                                                        float* __restrict__ tok_out) {
    // LDS: two 8KB buffers, float2 layout [k_pair][col], k_pair = k/2 (k even).
    __shared__ float2 sb[2][64 * 16];

    const int tid  = threadIdx.x;
    const int lane = tid & 31;
    const int col  = lane & 15;   // N column within 16-wide chunk / M row for A
    const int kh   = lane >> 4;   // half-wave selector (K offset 0 or 2)
    const int wave = tid >> 5;
    const int rowBase = blockIdx.x * 128 + wave * 16;

    // ---- A tile resident in registers: a[kk] = -2 * x[row][4kk + 2kh .. +1]
    v2f a[32];
    {
        const float* xrow = x + (size_t)(rowBase + col) * VQ_D + 2 * kh;
#pragma unroll
        for (int kk = 0; kk < 32; ++kk) {
            float2 t = *(const float2*)(xrow + 4 * kk);
            a[kk].x = -2.0f * t.x;
            a[kk].y = -2.0f * t.y;
        }
    }

    // Cooperative staging: thread t loads codebook row (chunk*16 + t&15),
    // k-range [(t>>4)*8, +8) as two float4, stores as four float2 LDS rows.
    const int scol = tid & 15;
    const int kp0  = (tid >> 4) * 4;  // k-pair base = (seg*8)/2

    auto stage_load = [&](int chunk, float4& g1, float4& g2) {
        const float* src = cb + (size_t)(chunk * 16 + scol) * VQ_D + (tid >> 4) * 8;
        g1 = *(const float4*)(src);
        g2 = *(const float4*)(src + 4);
    };
    auto stage_store = [&](int buf, float4 g1, float4 g2) {
        sb[buf][(kp0 + 0) * 16 + scol] = make_float2(g1.x, g1.y);
        sb[buf][(kp0 + 1) * 16 + scol] = make_float2(g1.z, g1.w);
        sb[buf][(kp0 + 2) * 16 + scol] = make_float2(g2.x, g2.y);
        sb[buf][(kp0 + 3) * 16 + scol] = make_float2(g2.z, g2.w);
    };

    {
        float4 g1, g2;
        stage_load(0, g1, g2);
        stage_store(0, g1, g2);
    }
    __syncthreads();

    float best[8];
    int   bidx[8];
#pragma unroll
    for (int r = 0; r < 8; ++r) { best[r] = 3.0e38f; bidx[r] = 0; }

    for (int c = 0; c < VQ_CHUNKS; ++c) {
        const int cur = c & 1;
        const bool more = (c + 1 < VQ_CHUNKS);
        float4 g1, g2;
        if (more) stage_load(c + 1, g1, g2);

        // Accumulators: acc[0] seeded with ||e_n||^2 (n = col, same for all 8
        // C elements of a lane); 4 accumulators break the WMMA RAW chain.
        const float es = esq[c * 16 + col];
        v8f acc[4];
#pragma unroll
        for (int r = 0; r < 8; ++r) {
            acc[0][r] = es; acc[1][r] = 0.0f; acc[2][r] = 0.0f; acc[3][r] = 0.0f;
        }

        const float2* bp = &sb[cur][kh * 16 + col];
#pragma unroll
        for (int kk = 0; kk < 32; ++kk) {
            float2 bb = bp[kk * 32];        // (2kk+kh)*16 + col
            v2f bv; bv.x = bb.x; bv.y = bb.y;
            acc[kk & 3] = __builtin_amdgcn_wmma_f32_16x16x4_f32(
                false, a[kk], false, bv, (short)0, acc[kk & 3], false, false);
        }

        v8f d = acc[0] + acc[1] + acc[2] + acc[3];
        const int nglob = c * 16 + col;
#pragma unroll
        for (int r = 0; r < 8; ++r) {
            float dv = d[r];
            if (dv < best[r]) { best[r] = dv; bidx[r] = nglob; }
        }

        if (more) stage_store(1 - cur, g1, g2);
        __syncthreads();
    }

    // Cross-lane argmin within each 16-lane half (C layout: VGPR r holds row
    // r (lanes 0-15) / row r+8 (lanes 16-31); N = lane&15). Tie -> lowest idx.
#pragma unroll
    for (int r = 0; r < 8; ++r) {
#pragma unroll
        for (int m = 8; m >= 1; m >>= 1) {
            float ov = __shfl_xor(best[r], m, 32);
            int   oi = __shfl_xor(bidx[r], m, 32);
            if (ov < best[r] || (ov == best[r] && oi < bidx[r])) {
                best[r] = ov; bidx[r] = oi;
            }
        }
    }
    if (col == 0) {
#pragma unroll
        for (int r = 0; r < 8; ++r)
            tok_out[rowBase + kh * 8 + r] = (float)bidx[r];
    }
}

// --------------------- gather codebook rows + loss partials -----------------
// 1024 blocks x 256 threads, 8192 elements/block, fixed-order double accum.
__global__ __launch_bounds__(256) void vq_gather_kernel(const float* __restrict__ x,
                                                        const float* __restrict__ cb,
                                                        const float* __restrict__ tok_f,
                                                        float* __restrict__ q_out,
                                                        double* __restrict__ partials) {
    __shared__ double red[256];
    const int tid = threadIdx.x;
    const size_t base = (size_t)blockIdx.x * 8192;
    double s = 0.0;
#pragma unroll 4
    for (int i = 0; i < 32; ++i) {
        size_t e = base + (size_t)i * 256 + tid;
        int tk = (int)tok_f[e >> 7];                       // exact for idx < 2^24
        float q = cb[(size_t)tk * VQ_D + (e & (VQ_D - 1))];
        q_out[e] = q;
        float dlt = q - x[e];
        s += (double)dlt * (double)dlt;
    }
    red[tid] = s;
    __syncthreads();
    for (int off = 128; off > 0; off >>= 1) {
        if (tid < off) red[tid] += red[tid + off];
        __syncthreads();
    }
    if (tid == 0) partials[blockIdx.x] = red[0];
}

__global__ __launch_bounds__(256) void vq_loss_kernel(const double* __restrict__ partials,
                                                      float* __restrict__ loss_out) {
    __shared__ double red[256];
    const int tid = threadIdx.x;
    double s = partials[tid] + partials[tid + 256] + partials[tid + 512] + partials[tid + 768];
    red[tid] = s;
    __syncthreads();
    for (int off = 128; off > 0; off >>= 1) {
        if (tid < off) red[tid] += red[tid + off];
        __syncthreads();
    }
    if (tid == 0)
        *loss_out = (float)(1.25 * red[0] / (double)VQ_ELEMS);  // (1+BETA)*mse
}

// --------------------------------- launcher ---------------------------------
extern "C" void kernel_launch(void* const* d_in, const int* in_sizes, int n_in,
                              void* d_out, int out_size, void* d_ws, size_t ws_size,
                              hipStream_t stream) {
    const float* x  = (const float*)d_in[0];   // [16, 4096, 128] f32
    const float* cb = (const float*)d_in[1];   // [4096, 128] f32

    float* out      = (float*)d_out;
    float* tok_out  = out;                       // 65536 tokens (as float)
    float* q_out    = out + VQ_BN;               // 8388608 quantized
    float* loss_out = out + VQ_BN + VQ_ELEMS;    // 1 scalar

    float*  esq      = (float*)d_ws;                                  // 16 KB
    double* partials = (double*)((char*)d_ws + VQ_V * sizeof(float)); // 8 KB

    vq_esq_kernel<<<VQ_V / 256, 256, 0, stream>>>(cb, esq);
    vq_argmin_kernel<<<VQ_BN / 128, 256, 0, stream>>>(x, cb, esq, tok_out);
    vq_gather_kernel<<<VQ_ELEMS / 8192, 256, 0, stream>>>(x, cb, tok_out, q_out, partials);
    vq_loss_kernel<<<1, 256, 0, stream>>>(partials, loss_out);
}